// SimGCNAlt_88914412962548
// MI455X (gfx1250) — compile-verified
//
#include <hip/hip_runtime.h>
#include <math.h>

typedef __attribute__((ext_vector_type(2))) float v2f;
typedef __attribute__((ext_vector_type(8))) float v8f;

#define TPB 256

static inline int cdiv(int a, int b) { return (a + b - 1) / b; }

// ---------------------------------------------------------------- fill
__global__ void fill_kernel(float* __restrict__ p, int n, float v) {
  int i = blockIdx.x * blockDim.x + threadIdx.x;
  if (i < n) p[i] = v;
}

// ------------------------------------------------------- degree (dst side)
__global__ void degree_kernel(const int* __restrict__ dst, int E,
                              float* __restrict__ deg) {
  int i = blockIdx.x * blockDim.x + threadIdx.x;
  int stride = gridDim.x * blockDim.x;
  for (; i < E; i += stride) atomicAdd(&deg[dst[i]], 1.0f);
}

__global__ void rsqrt_kernel(float* __restrict__ deg, int n) {
  int i = blockIdx.x * blockDim.x + threadIdx.x;
  if (i < n) deg[i] = rsqrtf(fmaxf(deg[i], 1.0f));
}

// ------------------------------------------- layer 1 transform (1 -> 4)
// g[n][c] = dinv[n] * (scale * x[n]) * W[0][c]
__global__ void layer1_kernel(const float* __restrict__ xin, float scale,
                              const float* __restrict__ W,  // [1][4]
                              const float* __restrict__ dinv,
                              float* __restrict__ g, int N) {
  int n = blockIdx.x * blockDim.x + threadIdx.x;
  if (n >= N) return;
  float v = xin[n] * scale * dinv[n];
  float4 r;
  r.x = v * W[0]; r.y = v * W[1]; r.z = v * W[2]; r.w = v * W[3];
  reinterpret_cast<float4*>(g)[n] = r;
}

// ------------------------------------------- layer 2 transform (4 -> 4), WMMA
// One wave computes a 16-node tile: D(16x16) = A(16x4 = activations) x B(4x16 = W
// zero-padded beyond col 3).  g[n][c] = dinv[n] * D[n][c], c in 0..3.
// A layout (f32 16x4): lanes 0-15 hold K=0,1 ; lanes 16-31 hold K=2,3.
// D layout: VGPR r, lanes 0-15 -> (M=r, N=lane), lanes 16-31 -> (M=8+r, N=lane-16).
// N (=100000) is a multiple of 16, so the guard-free full-tile path is the
// one that runs; the tail path exists only for generality.
__global__ __launch_bounds__(128)
void layer2_wmma_kernel(const float* __restrict__ act,  // [N][4]
                        const float* __restrict__ W,    // [4][4] row-major (k, c)
                        const float* __restrict__ dinv,
                        float* __restrict__ g, int N) {
  int wave = threadIdx.x >> 5;
  int lane = threadIdx.x & 31;
  int tile = blockIdx.x * 4 + wave;
  int base = tile * 16;
  if (base >= N) return;               // wave-uniform exit; EXEC all-1 at WMMA

  const bool full = (base + 16 <= N);  // wave-uniform

  int half = lane >> 4;                // 0: K=0,1   1: K=2,3
  int l    = lane & 15;
  int node = base + l;

  // A operand: node (base+l), two consecutive K values (one b64 load)
  v2f a;
  if (full || node < N) {
    const float* p = act + (size_t)node * 4 + half * 2;
    a.x = p[0]; a.y = p[1];
  } else {
    a.x = 0.0f; a.y = 0.0f;
  }

  // B operand: column l (valid for l<4), two consecutive K values
  v2f b;
  if (l < 4) {
    b.x = W[(2 * half + 0) * 4 + l];
    b.y = W[(2 * half + 1) * 4 + l];
  } else {
    b.x = 0.0f; b.y = 0.0f;
  }

  v8f c = {};
  v8f d = __builtin_amdgcn_wmma_f32_16x16x4_f32(
      /*neg_a=*/false, a, /*neg_b=*/false, b,
      /*c_mod=*/(short)0, c, /*reuse_a=*/false, /*reuse_b=*/false);

  if (l < 4) {
    int m0 = base + 8 * half;
    if (full) {
      // straight-line: 8 dinv loads + 8 stores, no per-element EXEC juggling
#pragma unroll
      for (int r = 0; r < 8; ++r) {
        g[(size_t)(m0 + r) * 4 + l] = dinv[m0 + r] * d[r];
      }
    } else {
#pragma unroll
      for (int r = 0; r < 8; ++r) {
        int m = m0 + r;
        if (m < N) g[(size_t)m * 4 + l] = dinv[m] * d[r];
      }
    }
  }
}

// ------------------------------------------- layer 3 transform (4 -> 1)
__global__ void layer3_kernel(const float* __restrict__ act,  // [N][4]
                              const float* __restrict__ W,    // [4][1]
                              const float* __restrict__ dinv,
                              float* __restrict__ gs, int N) {
  int n = blockIdx.x * blockDim.x + threadIdx.x;
  if (n >= N) return;
  float4 a = reinterpret_cast<const float4*>(act)[n];
  gs[n] = dinv[n] * (a.x * W[0] + a.y * W[1] + a.z * W[2] + a.w * W[3]);
}

// ------------------------------------------- edge scatter, 4-channel
__global__ void scatter4_kernel(const int* __restrict__ src,
                                const int* __restrict__ dst, int E,
                                const float* __restrict__ g,
                                float* __restrict__ agg) {
  int i = blockIdx.x * blockDim.x + threadIdx.x;
  int stride = gridDim.x * blockDim.x;
  for (; i < E; i += stride) {
    int s = src[i], t = dst[i];
    float4 v = reinterpret_cast<const float4*>(g)[s];
    atomicAdd(&agg[4 * t + 0], v.x);
    atomicAdd(&agg[4 * t + 1], v.y);
    atomicAdd(&agg[4 * t + 2], v.z);
    atomicAdd(&agg[4 * t + 3], v.w);
  }
}

// ------------------------------------------- edge scatter, scalar
__global__ void scatter1_kernel(const int* __restrict__ src,
                                const int* __restrict__ dst, int E,
                                const float* __restrict__ gs,
                                float* __restrict__ aggs) {
  int i = blockIdx.x * blockDim.x + threadIdx.x;
  int stride = gridDim.x * blockDim.x;
  for (; i < E; i += stride) atomicAdd(&aggs[dst[i]], gs[src[i]]);
}

// ------------------------------------------- finalize 4-wide (+bias, ReLU)
__global__ void finalize4_kernel(const float* __restrict__ agg,
                                 const float* __restrict__ g,
                                 const float* __restrict__ dinv,
                                 const float* __restrict__ b,
                                 float* __restrict__ out, int N) {
  int n = blockIdx.x * blockDim.x + threadIdx.x;
  if (n >= N) return;
  float di = dinv[n];
  float4 a = reinterpret_cast<const float4*>(agg)[n];
  float4 s = reinterpret_cast<const float4*>(g)[n];
  float4 r;
  r.x = fmaxf(di * (a.x + s.x) + b[0], 0.0f);
  r.y = fmaxf(di * (a.y + s.y) + b[1], 0.0f);
  r.z = fmaxf(di * (a.z + s.z) + b[2], 0.0f);
  r.w = fmaxf(di * (a.w + s.w) + b[3], 0.0f);
  reinterpret_cast<float4*>(out)[n] = r;
}

// ------------------------------------------- finalize scalar (optionally sin(10*))
__global__ void finalize_s_kernel(const float* __restrict__ aggs,
                                  const float* __restrict__ gs,
                                  const float* __restrict__ dinv,
                                  const float* __restrict__ b,
                                  float* __restrict__ out, int N, int apply_sin) {
  int n = blockIdx.x * blockDim.x + threadIdx.x;
  if (n >= N) return;
  float v = dinv[n] * (aggs[n] + gs[n]) + b[0];
  out[n] = apply_sin ? sinf(10.0f * v) : v;
}

// ------------------------------------------- node embeddings + mean accumulation
__global__ void emb_reduce_kernel(const float* __restrict__ x2,
                                  const float* __restrict__ x3,
                                  float* __restrict__ out,
                                  float* __restrict__ accum, int N) {
  __shared__ float s2[TPB], s3[TPB];
  int tid = threadIdx.x;
  float a = 0.0f, b = 0.0f;
  for (int n = blockIdx.x * blockDim.x + tid; n < N;
       n += gridDim.x * blockDim.x) {
    float v2 = x2[n], v3 = x3[n];
    out[2 + 2 * n]     = v2;
    out[2 + 2 * n + 1] = v3;
    a += v2; b += v3;
  }
  s2[tid] = a; s3[tid] = b;
  __syncthreads();
  for (int off = TPB / 2; off > 0; off >>= 1) {
    if (tid < off) { s2[tid] += s2[tid + off]; s3[tid] += s3[tid + off]; }
    __syncthreads();
  }
  if (tid == 0) {
    atomicAdd(&accum[0], s2[0]);
    atomicAdd(&accum[1], s3[0]);
  }
}

__global__ void graphemb_kernel(const float* __restrict__ accum,
                                float* __restrict__ out, int N) {
  if (threadIdx.x == 0) {
    float inv = 1.0f / (float)N;
    out[0] = accum[0] * inv;
    out[1] = accum[1] * inv;
  }
}

// ----------------------------------------------------------------------------
extern "C" void kernel_launch(void* const* d_in, const int* in_sizes, int n_in,
                              void* d_out, int out_size, void* d_ws,
                              size_t ws_size, hipStream_t stream) {
  const float* x   = (const float*)d_in[0];
  const int*   ei  = (const int*)d_in[1];
  const float* w11 = (const float*)d_in[2];
  const float* b11 = (const float*)d_in[3];
  const float* w12 = (const float*)d_in[4];
  const float* b12 = (const float*)d_in[5];
  const float* w13 = (const float*)d_in[6];
  const float* b13 = (const float*)d_in[7];
  const float* w21 = (const float*)d_in[8];
  const float* b21 = (const float*)d_in[9];
  const float* w22 = (const float*)d_in[10];
  const float* b22 = (const float*)d_in[11];
  const float* w23 = (const float*)d_in[12];
  const float* b23 = (const float*)d_in[13];

  const int N = in_sizes[0];      // 100000
  const int E = in_sizes[1] / 2;  // 6400000
  const int* src = ei;
  const int* dst = ei + E;

  float* ws    = (float*)d_ws;     // 17N + 2 floats used (~6.8 MB)
  float* g4    = ws;
  float* agg4  = ws + 4 * (size_t)N;
  float* act   = ws + 8 * (size_t)N;
  float* dinv  = ws + 12 * (size_t)N;
  float* gs    = ws + 13 * (size_t)N;
  float* aggs  = ws + 14 * (size_t)N;
  float* x2    = ws + 15 * (size_t)N;
  float* x3    = ws + 16 * (size_t)N;
  float* accum = ws + 17 * (size_t)N;

  float* out = (float*)d_out;

  const int gN  = cdiv(N, TPB);
  const int g4N = cdiv(4 * N, TPB);
  int gE = cdiv(E, TPB);
  if (gE > 16384) gE = 16384;               // grid-stride over edges
  const int gT = cdiv(cdiv(N, 16), 4);      // 4 waves / block, 16 nodes / wave

  // ---- normalization: deg = 1 (self loop) + indegree ; dinv = rsqrt(max(deg,1))
  fill_kernel<<<gN, TPB, 0, stream>>>(dinv, N, 1.0f);
  degree_kernel<<<gE, TPB, 0, stream>>>(dst, E, dinv);
  rsqrt_kernel<<<gN, TPB, 0, stream>>>(dinv, N);

  // ==================== GCN stack #1 (input x * 10) ====================
  layer1_kernel<<<gN, TPB, 0, stream>>>(x, 10.0f, w11, dinv, g4, N);
  fill_kernel<<<g4N, TPB, 0, stream>>>(agg4, 4 * N, 0.0f);
  scatter4_kernel<<<gE, TPB, 0, stream>>>(src, dst, E, g4, agg4);
  finalize4_kernel<<<gN, TPB, 0, stream>>>(agg4, g4, dinv, b11, act, N);

  layer2_wmma_kernel<<<gT, 128, 0, stream>>>(act, w12, dinv, g4, N);
  fill_kernel<<<g4N, TPB, 0, stream>>>(agg4, 4 * N, 0.0f);
  scatter4_kernel<<<gE, TPB, 0, stream>>>(src, dst, E, g4, agg4);
  finalize4_kernel<<<gN, TPB, 0, stream>>>(agg4, g4, dinv, b12, act, N);

  layer3_kernel<<<gN, TPB, 0, stream>>>(act, w13, dinv, gs, N);
  fill_kernel<<<gN, TPB, 0, stream>>>(aggs, N, 0.0f);
  scatter1_kernel<<<gE, TPB, 0, stream>>>(src, dst, E, gs, aggs);
  finalize_s_kernel<<<gN, TPB, 0, stream>>>(aggs, gs, dinv, b13, x2, N, 1);

  // ==================== GCN stack #2 (input x2) ========================
  layer1_kernel<<<gN, TPB, 0, stream>>>(x2, 1.0f, w21, dinv, g4, N);
  fill_kernel<<<g4N, TPB, 0, stream>>>(agg4, 4 * N, 0.0f);
  scatter4_kernel<<<gE, TPB, 0, stream>>>(src, dst, E, g4, agg4);
  finalize4_kernel<<<gN, TPB, 0, stream>>>(agg4, g4, dinv, b21, act, N);

  layer2_wmma_kernel<<<gT, 128, 0, stream>>>(act, w22, dinv, g4, N);
  fill_kernel<<<g4N, TPB, 0, stream>>>(agg4, 4 * N, 0.0f);
  scatter4_kernel<<<gE, TPB, 0, stream>>>(src, dst, E, g4, agg4);
  finalize4_kernel<<<gN, TPB, 0, stream>>>(agg4, g4, dinv, b22, act, N);

  layer3_kernel<<<gN, TPB, 0, stream>>>(act, w23, dinv, gs, N);
  fill_kernel<<<gN, TPB, 0, stream>>>(aggs, N, 0.0f);
  scatter1_kernel<<<gE, TPB, 0, stream>>>(src, dst, E, gs, aggs);
  finalize_s_kernel<<<gN, TPB, 0, stream>>>(aggs, gs, dinv, b23, x3, N, 0);

  // ==================== outputs ========================================
  fill_kernel<<<1, TPB, 0, stream>>>(accum, 2, 0.0f);
  emb_reduce_kernel<<<gN, TPB, 0, stream>>>(x2, x3, out, accum, N);
  graphemb_kernel<<<1, 32, 0, stream>>>(accum, out, N);
}